// DagnabbitAutoEncoder_74062416052349
// MI455X (gfx1250) — compile-verified
//
#include <hip/hip_runtime.h>
#include <math.h>

typedef __attribute__((ext_vector_type(2))) float v2f;
typedef __attribute__((ext_vector_type(4))) float v4f;
typedef __attribute__((ext_vector_type(8))) float v8f;

#define D_        128
#define R_        1024
#define T_        4
#define OUTSLOTS_ 256
#define START_    4
#define L_        64
#define M_        2048
#define N_        (R_ + L_ * M_)
#define NENC_     5          // T+1 encoders
#define K1_       256        // 2*D, inner dim of both layers
#define XSTR_     260        // padded LDS row stride (floats); 260 % 64 == 4 -> conflict-free

// packed-weight geometry: fragment index ((e*NT + j)*64 + k4)*32 + lane, 2 floats each
#define W1P_FRAGS (NENC_ * 16 * 64 * 32)             // 163840 fragments
#define W2P_FRAGS (NENC_ * 8  * 64 * 32)             // 81920 fragments
#define W1P_BYTES ((size_t)W1P_FRAGS * 8)            // 1310720
#define W2P_BYTES ((size_t)W2P_FRAGS * 8)            // 655360
#define WSP_BYTES (W1P_BYTES + W2P_BYTES)            // 1966080

// ---------------------------------------------------------------------------
// Init: copy root embeddings into rows [0, R) of the node buffer.
// ---------------------------------------------------------------------------
__global__ void dag_init_roots(float* __restrict__ buf, const float* __restrict__ root) {
    int i = blockIdx.x * blockDim.x + threadIdx.x;     // one float4 per thread
    v4f v = *(const v4f*)(root + (size_t)i * 4);
    *(v4f*)(buf + (size_t)i * 4) = v;
}

// ---------------------------------------------------------------------------
// Pack weights into WMMA 4x16 B-fragment lane order:
//   dst[frag] = { W[kk][col], W[kk+1][col] },  kk = k4*4 + 2*(lane>>4), col = lane&15
// so the inner loop does ONE coalesced global_load_b64 per WMMA.
// ---------------------------------------------------------------------------
__global__ void dag_pack_w(const float* __restrict__ W, float* __restrict__ Wp,
                           int ntile, int ncol) {     // ncol = output dim (256 or 128)
    int idx  = blockIdx.x * blockDim.x + threadIdx.x; // one fragment per thread
    int lane = idx & 31;
    int k4   = (idx >> 5) & 63;
    int j    = (idx >> 11) % ntile;
    int e    = idx / (ntile << 11);
    int col  = lane & 15;
    int kk   = k4 * 4 + ((lane >> 4) & 1) * 2;
    const float* Wsrc = W + (size_t)e * K1_ * ncol + (size_t)(j * 16 + col);
    Wp[(size_t)idx * 2 + 0] = Wsrc[(size_t)kk * ncol];
    Wp[(size_t)idx * 2 + 1] = Wsrc[(size_t)(kk + 1) * ncol];
}

// ---------------------------------------------------------------------------
// One topological level. 5 waves per block (one per encoder e), one 16-node
// tile per block; 128 blocks. Row-type selection == disjoint write ownership.
// ---------------------------------------------------------------------------
template <bool PACKED>
__global__ __launch_bounds__(160)
void dag_level(float* __restrict__ buf,            // (N, D) node buffer == d_out
               const float* __restrict__ W1,       // raw (5,256,256) or packed frags
               const float* __restrict__ b1,       // (5, 256)
               const float* __restrict__ W2,       // raw (5,256,128) or packed frags
               const float* __restrict__ b2,       // (5, 128)
               const float* __restrict__ slots,    // (256, 128)
               const int*   __restrict__ par,      // (N-R, 2)
               const int*   __restrict__ types,    // (N-R)
               int level)
{
    __shared__ float X[16 * XSTR_];   // layer-1 input panel, 16 x 256 (padded)
    __shared__ float H[16 * XSTR_];   // layer-2 input panel (selected gelu output)

    const int lane     = threadIdx.x & 31;
    const int e        = threadIdx.x >> 5;       // encoder handled by this wave, 0..4
    const int tile     = blockIdx.x;             // 0..127
    const int qbase    = level * M_ + tile * 16; // index into par/types
    const int rowbase  = R_ + qbase;             // global output row of tile row 0
    const int colLane  = lane & 15;
    const int laneHalf = (lane >> 4) & 1;
    const int klo      = laneHalf * 2;           // K sub-offset for 16x16x4 f32 layout

    // ---- cooperative gather: wave e handles rows m = e, e+5, e+10, ... ----
    for (int m = e; m < 16; m += NENC_) {
        int p0 = par[(qbase + m) * 2 + 0];
        int p1 = par[(qbase + m) * 2 + 1];
        int t  = types[qbase + m];
        const float* s0 = buf + (size_t)p0 * D_;
        const float* s1;
        if (t >= START_) {
            int slot = t - START_;
            slot = slot < 0 ? 0 : (slot > OUTSLOTS_ - 1 ? OUTSLOTS_ - 1 : slot);
            s1 = slots + (size_t)slot * D_;
        } else {
            s1 = buf + (size_t)p1 * D_;
        }
        v4f a = *(const v4f*)(s0 + lane * 4);    // 32 lanes x 4 floats = 128
        v4f c = *(const v4f*)(s1 + lane * 4);
        *(v4f*)(&X[m * XSTR_ + lane * 4])       = a;
        *(v4f*)(&X[m * XSTR_ + 128 + lane * 4]) = c;
    }

    // per-lane row ownership: row r of the C/D tile -> encoder index
    int rt[8];
    bool ownAny = false;
    #pragma unroll
    for (int i = 0; i < 8; ++i) {
        int t = types[qbase + i + laneHalf * 8];
        rt[i] = (t >= START_) ? T_ : t;
        ownAny |= (rt[i] == e);
    }
    // wave-uniform: does this wave own ANY of the 16 rows of this tile?
    const bool waveActive = __any(ownAny);
    __syncthreads();

    // ---- layer 1: x @ W1[e] + b1[e], gelu, store owned rows into H ----
    if (waveActive) {
        #pragma unroll 1
        for (int j = 0; j < 16; ++j) {           // output N-tile (16 cols each)
            v8f acc = {};
            if (PACKED) {
                const v2f* Wf = (const v2f*)W1 + ((size_t)(e * 16 + j) * 64) * 32 + lane;
                #pragma unroll 8
                for (int k4 = 0; k4 < 64; ++k4) {
                    int kk = k4 * 4 + klo;
                    v2f a = *(const v2f*)(&X[colLane * XSTR_ + kk]);
                    v2f b = Wf[k4 * 32];         // one coalesced global_load_b64
                    acc = __builtin_amdgcn_wmma_f32_16x16x4_f32(
                            false, a, false, b, (short)0, acc, false, false);
                }
            } else {
                const float* Wb = W1 + (size_t)e * K1_ * K1_ + (size_t)(j * 16 + colLane);
                #pragma unroll 8
                for (int k4 = 0; k4 < 64; ++k4) {
                    int kk = k4 * 4 + klo;
                    v2f a = *(const v2f*)(&X[colLane * XSTR_ + kk]);
                    v2f b;
                    b.x = Wb[(size_t)kk * K1_];
                    b.y = Wb[(size_t)(kk + 1) * K1_];
                    acc = __builtin_amdgcn_wmma_f32_16x16x4_f32(
                            false, a, false, b, (short)0, acc, false, false);
                }
            }
            float bias = b1[e * K1_ + j * 16 + colLane];
            #pragma unroll
            for (int i = 0; i < 8; ++i) {
                // ownership is uniform per half-wave -> execz-skips whole GELU
                if (rt[i] == e) {
                    float x = acc[i] + bias;
                    float g = 0.5f * x * (1.0f + erff(x * 0.70710678118654752f));
                    H[(i + laneHalf * 8) * XSTR_ + j * 16 + colLane] = g;
                }
            }
        }
    }
    __syncthreads();

    // ---- layer 2: h @ W2[e] + b2[e], store owned rows straight to the buffer ----
    if (waveActive) {
        #pragma unroll 1
        for (int j = 0; j < 8; ++j) {            // output N-tile over D=128
            v8f acc = {};
            if (PACKED) {
                const v2f* Wf = (const v2f*)W2 + ((size_t)(e * 8 + j) * 64) * 32 + lane;
                #pragma unroll 8
                for (int k4 = 0; k4 < 64; ++k4) {
                    int kk = k4 * 4 + klo;
                    v2f a = *(const v2f*)(&H[colLane * XSTR_ + kk]);
                    v2f b = Wf[k4 * 32];
                    acc = __builtin_amdgcn_wmma_f32_16x16x4_f32(
                            false, a, false, b, (short)0, acc, false, false);
                }
            } else {
                const float* Wb = W2 + (size_t)e * K1_ * D_ + (size_t)(j * 16 + colLane);
                #pragma unroll 8
                for (int k4 = 0; k4 < 64; ++k4) {
                    int kk = k4 * 4 + klo;
                    v2f a = *(const v2f*)(&H[colLane * XSTR_ + kk]);
                    v2f b;
                    b.x = Wb[(size_t)kk * D_];
                    b.y = Wb[(size_t)(kk + 1) * D_];
                    acc = __builtin_amdgcn_wmma_f32_16x16x4_f32(
                            false, a, false, b, (short)0, acc, false, false);
                }
            }
            float bias = b2[e * D_ + j * 16 + colLane];
            #pragma unroll
            for (int i = 0; i < 8; ++i) {
                if (rt[i] == e) {
                    int row = rowbase + i + laneHalf * 8;
                    buf[(size_t)row * D_ + j * 16 + colLane] = acc[i] + bias;
                }
            }
        }
    }
}

// ---------------------------------------------------------------------------
extern "C" void kernel_launch(void* const* d_in, const int* in_sizes, int n_in,
                              void* d_out, int out_size, void* d_ws, size_t ws_size,
                              hipStream_t stream) {
    const float* root  = (const float*)d_in[0];  // (R, D)
    const float* W1    = (const float*)d_in[1];  // (5, 256, 256)
    const float* b1    = (const float*)d_in[2];  // (5, 256)
    const float* W2    = (const float*)d_in[3];  // (5, 256, 128)
    const float* b2    = (const float*)d_in[4];  // (5, 128)
    const float* slots = (const float*)d_in[5];  // (256, 128)
    const int*   par   = (const int*)d_in[6];    // (N-R, 2)
    const int*   typ   = (const int*)d_in[7];    // (N-R)
    float* out = (float*)d_out;                  // (N, D) — fully overwritten

    // roots: R*D = 131072 floats = 32768 float4
    dag_init_roots<<<128, 256, 0, stream>>>(out, root);

    const bool packed = (ws_size >= WSP_BYTES);  // constant per run -> deterministic
    float* W1p = (float*)d_ws;
    float* W2p = (float*)((char*)d_ws + W1P_BYTES);

    if (packed) {
        dag_pack_w<<<W1P_FRAGS / 256, 256, 0, stream>>>(W1, W1p, 16, K1_);
        dag_pack_w<<<W2P_FRAGS / 256, 256, 0, stream>>>(W2, W2p, 8, D_);
        for (int l = 0; l < L_; ++l)
            dag_level<true><<<128, 160, 0, stream>>>(out, W1p, b1, W2p, b2,
                                                     slots, par, typ, l);
    } else {
        for (int l = 0; l < L_; ++l)
            dag_level<false><<<128, 160, 0, stream>>>(out, W1, b1, W2, b2,
                                                      slots, par, typ, l);
    }
}